// GCNEncoder_20822001451037
// MI455X (gfx1250) — compile-verified
//
#include <hip/hip_runtime.h>

typedef __attribute__((ext_vector_type(2))) float v2f;
typedef __attribute__((ext_vector_type(4))) float v4f;
typedef __attribute__((ext_vector_type(8))) float v8f;

#define D 128  // D_IN == D_H == 128

// ---------------- kernel 1: zero out-accumulator, deg = 1 (self loop) ----------------
__global__ void gcn_init(float* __restrict__ out, float* __restrict__ deg,
                         int nOut, int nNodes) {
  int i = blockIdx.x * blockDim.x + threadIdx.x;
  if (i < nOut)   out[i] = 0.0f;
  if (i < nNodes) deg[i] = 1.0f;
}

// ---------------- kernel 2: deg[dst] += 1 per edge (HW f32 atomic) ----------------
__global__ void gcn_degree(const int* __restrict__ dst, float* __restrict__ deg,
                           int nEdges) {
  int e = blockIdx.x * blockDim.x + threadIdx.x;
  if (e < nEdges) unsafeAtomicAdd(&deg[dst[e]], 1.0f);
}

// ---------------- kernel 3: hs = (X @ W) * rsqrt(deg)  via V_WMMA_F32_16X16X4_F32 ----
// Block = 128 threads = 4 waves. Each wave computes a 16-row x 128-col strip.
// W (128x128 f32 = 64KB) staged in LDS, pair-packed over K so each B fragment
// is one ds_load_b64: element (k,n) lives at sW[(k>>1)*256 + n*2 + (k&1)].
__global__ void __launch_bounds__(128)
gcn_gemm(const float* __restrict__ X, const float* __restrict__ W,
         const float* __restrict__ deg, float* __restrict__ hs, int nRows)
{
  __shared__ float sW[(D / 2) * 256];  // 64 KB

  const int tid  = threadIdx.x;
  const int wave = tid >> 5;
  const int lane = tid & 31;
  const int half = lane >> 4;   // 0: lanes 0-15, 1: lanes 16-31
  const int l16  = lane & 15;

  // Stage W into LDS (coalesced global reads, pair-packed LDS writes)
  for (int i = tid; i < D * D; i += 128) {
    int k = i >> 7;
    int n = i & (D - 1);
    sW[(k >> 1) * 256 + n * 2 + (k & 1)] = W[i];
  }
  __syncthreads();

  const int rowBase = (blockIdx.x * 4 + wave) * 16;

  // A-matrix 16x4 f32 layout: lane = M (mod 16), VGPR r holds K = r + 2*half.
  // Clamp OOB rows (select, not branch: EXEC must stay all-1s for WMMA).
  int arow = rowBase + l16;
  if (arow >= nRows) arow = nRows - 1;
  const float* Xrow = X + (size_t)arow * D;

  v8f acc[8] = {};

  for (int k0 = 0; k0 < D; k0 += 4) {
    // A fragment: K = {k0+2*half, k0+1+2*half} -> contiguous b64 load
    v2f a = *(const v2f*)(Xrow + k0 + 2 * half);
    const float* sWk = sW + ((k0 + 2 * half) >> 1) * 256;
#pragma unroll
    for (int t = 0; t < 8; ++t) {
      // B fragment 4x16: lane = N (mod 16), VGPR r holds K = r + 2*half
      v2f b = *(const v2f*)(sWk + (t * 16 + l16) * 2);
      acc[t] = __builtin_amdgcn_wmma_f32_16x16x4_f32(
          false, a, false, b, (short)0, acc[t], false, false);
    }
  }

  // C/D 16x16 f32 layout: VGPR r -> M = r + 8*half, N = l16 (+16*t)
#pragma unroll
  for (int r = 0; r < 8; ++r) {
    int row = rowBase + r + 8 * half;
    if (row < nRows) {
      float dis  = rsqrtf(deg[row]);
      float* hrow = hs + (size_t)row * D + l16;
#pragma unroll
      for (int t = 0; t < 8; ++t)
        hrow[t * 16] = acc[t][r] * dis;
    }
  }
}

// ---------------- kernel 4: out[dst] += hs[src]  (one wave per edge) ----------------
__global__ void __launch_bounds__(256)
gcn_scatter(const int* __restrict__ src, const int* __restrict__ dst,
            const float* __restrict__ hs, float* __restrict__ out, int nEdges)
{
  int gid  = blockIdx.x * blockDim.x + threadIdx.x;
  int edge = gid >> 5;
  if (edge >= nEdges) return;
  int lane = gid & 31;
  int s = src[edge];
  int d = dst[edge];
  v4f h = *(const v4f*)(hs + (size_t)s * D + lane * 4);   // L2-resident gather
  float* o = out + (size_t)d * D + lane * 4;
  unsafeAtomicAdd(o + 0, h.x);
  unsafeAtomicAdd(o + 1, h.y);
  unsafeAtomicAdd(o + 2, h.z);
  unsafeAtomicAdd(o + 3, h.w);
}

// ---------------- kernel 5: out = relu(dis * (out + hs) + b) ----------------
__global__ void gcn_finish(float* __restrict__ out, const float* __restrict__ hs,
                           const float* __restrict__ deg, const float* __restrict__ bias,
                           int total)
{
  int i = blockIdx.x * blockDim.x + threadIdx.x;
  if (i >= total) return;
  int row = i >> 7;
  int col = i & (D - 1);
  float dis = rsqrtf(deg[row]);
  float v = fmaf(dis, out[i] + hs[i], bias[col]);
  out[i] = fmaxf(v, 0.0f);
}

extern "C" void kernel_launch(void* const* d_in, const int* in_sizes, int n_in,
                              void* d_out, int out_size, void* d_ws, size_t ws_size,
                              hipStream_t stream) {
  const float* X    = (const float*)d_in[0];  // [N, 128]
  const int*   eidx = (const int*)d_in[1];    // [2, E] row-major: src then dst
  const float* W    = (const float*)d_in[2];  // [128, 128]
  const float* b    = (const float*)d_in[3];  // [128]

  const int nNodes = in_sizes[0] / D;
  const int nEdges = in_sizes[1] / 2;
  const int* srcI = eidx;
  const int* dstI = eidx + nEdges;

  float* out = (float*)d_out;
  float* deg = (float*)d_ws;          // N floats
  float* hs  = deg + nNodes;          // N*128 floats (offset 400000 B, 16B-aligned)

  const int total = nNodes * D;

  gcn_init  <<<(total  + 255) / 256, 256, 0, stream>>>(out, deg, total, nNodes);
  gcn_degree<<<(nEdges + 255) / 256, 256, 0, stream>>>(dstI, deg, nEdges);
  gcn_gemm  <<<(nNodes + 63)  / 64,  128, 0, stream>>>(X, W, deg, hs, nNodes);

  long sthreads = (long)nEdges * 32;
  gcn_scatter<<<(int)((sthreads + 255) / 256), 256, 0, stream>>>(srcI, dstI, hs, out, nEdges);

  gcn_finish<<<(total + 255) / 256, 256, 0, stream>>>(out, hs, deg, b, total);
}